// GatedMDNLoss_88098369175988
// MI455X (gfx1250) — compile-verified
//
#include <hip/hip_runtime.h>
#include <hip/hip_bf16.h>
#include <math.h>

typedef __attribute__((ext_vector_type(2))) float v2f;
typedef __attribute__((ext_vector_type(8))) float v8f;

#define KCOMP 64
#define DDIM  8
#define NCPK  36
#define FDIM  48            // 36 quad + 8 linear + 1 const + 3 pad
#define TILES_PER_BLK 8     // 8 x 16 = 128 batch rows per block
#define NORMC (-7.3515082656373808f)   // -0.5*8*log(2*pi)

// np.tril_indices(8) order
__constant__ int c_ti[NCPK] = {0,1,1,2,2,2,3,3,3,3,4,4,4,4,4,5,5,5,5,5,5,
                               6,6,6,6,6,6,6,7,7,7,7,7,7,7,7};
__constant__ int c_tj[NCPK] = {0,0,1,0,1,2,0,1,2,3,0,1,2,3,4,0,1,2,3,4,5,
                               0,1,2,3,4,5,6,0,1,2,3,4,5,6,7};

// ---------------------------------------------------------------------------
// Kernel 1: per-component precompute.  W[k][0..47] such that
//   logprob[b,k] = sum_m W[k][m] * f_b[m]
// ---------------------------------------------------------------------------
__global__ void gmdn_prep_kernel(const float* __restrict__ mu,
                                 const float* __restrict__ Lc,
                                 float* __restrict__ W)
{
    int k = threadIdx.x;
    if (k >= KCOMP) return;

    float Ld[DDIM][DDIM];
    float Li[DDIM][DDIM];
    #pragma unroll
    for (int i = 0; i < DDIM; ++i)
        #pragma unroll
        for (int j = 0; j < DDIM; ++j) { Ld[i][j] = 0.f; Li[i][j] = 0.f; }

    for (int m = 0; m < NCPK; ++m)
        Ld[c_ti[m]][c_tj[m]] = Lc[k * NCPK + m];

    // invert lower-triangular L by forward substitution (column-wise)
    for (int c = 0; c < DDIM; ++c) {
        Li[c][c] = 1.0f / Ld[c][c];
        for (int r = c + 1; r < DDIM; ++r) {
            float s = 0.f;
            for (int m = c; m < r; ++m) s += Ld[r][m] * Li[m][c];
            Li[r][c] = -s / Ld[r][r];
        }
    }

    // P = Li^T * Li  (symmetric precision matrix)
    float P[DDIM][DDIM];
    for (int i = 0; i < DDIM; ++i)
        for (int j = 0; j < DDIM; ++j) {
            float s = 0.f;
            int m0 = i > j ? i : j;
            for (int m = m0; m < DDIM; ++m) s += Li[m][i] * Li[m][j];
            P[i][j] = s;
        }

    float muk[DDIM];
    #pragma unroll
    for (int i = 0; i < DDIM; ++i) muk[i] = mu[k * DDIM + i];

    float q[DDIM];
    float c0 = 0.f;
    for (int i = 0; i < DDIM; ++i) {
        float s = 0.f;
        for (int j = 0; j < DDIM; ++j) s += P[i][j] * muk[j];
        q[i] = s;
        c0 += s * muk[i];
    }

    float logdet = 0.f;
    #pragma unroll
    for (int i = 0; i < DDIM; ++i) logdet += __logf(Ld[i][i]);

    // logprob = NORMC - logdet - 0.5*(t^T P t) + q.t - 0.5*c0
    float* wr = W + k * FDIM;
    for (int m = 0; m < NCPK; ++m) {
        int i = c_ti[m], j = c_tj[m];
        wr[m] = (i == j) ? -0.5f * P[i][j] : -P[i][j];   // off-diag doubled
    }
    #pragma unroll
    for (int i = 0; i < DDIM; ++i) wr[NCPK + i] = q[i];
    wr[44] = NORMC - logdet - 0.5f * c0;
    wr[45] = 0.f; wr[46] = 0.f; wr[47] = 0.f;
}

// ---------------------------------------------------------------------------
// Kernel 2: each block handles 128 batch rows = 8 tiles of 16; W loaded to
// LDS once per block.  Per tile: coalesced pi+target staging, feature build,
// 12 x v_wmma_f32_16x16x4_f32 per wave (4 waves = 4 component tiles),
// +log(pi), per-row logsumexp over all 64 components.
// ---------------------------------------------------------------------------
__global__ void __launch_bounds__(128)
gmdn_main_kernel(const float* __restrict__ pi,
                 const float* __restrict__ target,
                 const float* __restrict__ W,
                 float* __restrict__ partials)
{
    __shared__ float sW[KCOMP][50];     // weights, padded stride
    __shared__ float sT[16 * DDIM];     // target tile
    __shared__ float sF[16][50];        // features, padded stride
    __shared__ float sPC[16][68];       // pi tile, then weighted logprobs (in place)
    __shared__ float sLse[16];

    const int tid = threadIdx.x;        // 0..127
    const long blockBase = (long)blockIdx.x * (16 * TILES_PER_BLK);

    // load 64x48 weight matrix once per block
    for (int i = tid; i < KCOMP * FDIM; i += 128)
        sW[i / FDIM][i % FDIM] = W[i];

    const int wave = tid >> 5;          // N-tile: components [wave*16, wave*16+16)
    const int lane = tid & 31;
    const int hi   = lane >> 4;         // 0: K{0,1}/rows 0-7 ; 1: K{2,3}/rows 8-15
    const int l16  = lane & 15;
    const int koff = hi * 2;
    const int col  = wave * 16 + l16;
    const int prow = tid >> 3;          // pi staging: row 0..15
    const int pcol = (tid & 7) * 8;     // pi staging: col base

    const float* frow = &sF[l16][0];              // A row (M = l16)
    const float* wrow = &sW[wave * 16 + l16][0];  // B column (N = col)

    float lseAcc = 0.f;                 // per-row LSE accumulator (tid < 16)

    for (int t = 0; t < TILES_PER_BLK; ++t) {
        const long rowBase = blockBase + t * 16;

        // coalesced staging: 16x8 target tile + 16x64 pi tile
        sT[tid] = target[rowBase * DDIM + tid];
        {
            const float4* psrc =
                reinterpret_cast<const float4*>(pi + rowBase * KCOMP) + tid * 2;
            float4 p0 = psrc[0];
            float4 p1 = psrc[1];
            *(float4*)&sPC[prow][pcol]     = p0;
            *(float4*)&sPC[prow][pcol + 4] = p1;
        }
        __syncthreads();

        // prefetch next tile's streams while we compute (global_prefetch_b8)
        if (t + 1 < TILES_PER_BLK) {
            const long nextBase = rowBase + 16;
            __builtin_prefetch(pi + nextBase * KCOMP + tid * 8, 0, 1);
            __builtin_prefetch(target + nextBase * DDIM + tid, 0, 1);
        }

        // build features: f = [t_i*t_j (36), t_i (8), 1, 0,0,0]
        for (int e = tid; e < 16 * FDIM; e += 128) {
            int r = e / FDIM, m = e % FDIM;
            float v;
            if (m < NCPK)          v = sT[r * DDIM + c_ti[m]] * sT[r * DDIM + c_tj[m]];
            else if (m < NCPK + 8) v = sT[r * DDIM + (m - NCPK)];
            else if (m == 44)      v = 1.0f;
            else                   v = 0.0f;
            sF[r][m] = v;
        }
        __syncthreads();

        v8f acc = {};
        #pragma unroll
        for (int kk = 0; kk < FDIM; kk += 4) {
            v2f a, b;
            a.x = frow[kk + koff];
            a.y = frow[kk + koff + 1];
            b.x = wrow[kk + koff];
            b.y = wrow[kk + koff + 1];
            // D = A(16x4, f32) * B(4x16, f32) + C
            acc = __builtin_amdgcn_wmma_f32_16x16x4_f32(
                false, a, false, b, (short)0, acc, false, false);
        }

        // weighted[b,k] = logprob + log(pi + 1e-10), in place over the pi tile.
        // C/D layout: VGPR r -> row r + hi*8.  Each cell is read+written by
        // exactly one lane, so no barrier needed between read and write.
        #pragma unroll
        for (int r = 0; r < 8; ++r) {
            int rowl = r + hi * 8;
            float p = sPC[rowl][col];
            sPC[rowl][col] = acc[r] + __logf(p + 1e-10f);
        }
        __syncthreads();

        // per-row logsumexp over 64 components
        if (tid < 16) {
            float mx = -INFINITY;
            for (int c = 0; c < KCOMP; ++c) mx = fmaxf(mx, sPC[tid][c]);
            float s = 0.f;
            for (int c = 0; c < KCOMP; ++c) s += __expf(sPC[tid][c] - mx);
            lseAcc += mx + __logf(s);
        }
        __syncthreads();   // protect sPC/sT before next tile overwrites
    }

    if (tid < 16) sLse[tid] = lseAcc;
    __syncthreads();
    if (tid == 0) {
        float s = 0.f;
        #pragma unroll
        for (int r = 0; r < 16; ++r) s += sLse[r];
        partials[blockIdx.x] = s;            // fixed order -> deterministic
    }
}

// ---------------------------------------------------------------------------
// Kernel 3: deterministic tree reduction of block partials -> nll
// ---------------------------------------------------------------------------
__global__ void gmdn_reduce_kernel(const float* __restrict__ partials, int n,
                                   float invB, float* __restrict__ out)
{
    __shared__ float s[256];
    float acc = 0.f;
    for (int i = threadIdx.x; i < n; i += 256) acc += partials[i];
    s[threadIdx.x] = acc;
    __syncthreads();
    for (int st = 128; st > 0; st >>= 1) {
        if (threadIdx.x < st) s[threadIdx.x] += s[threadIdx.x + st];
        __syncthreads();
    }
    if (threadIdx.x == 0) out[0] = -s[0] * invB;
}

// ---------------------------------------------------------------------------
extern "C" void kernel_launch(void* const* d_in, const int* in_sizes, int n_in,
                              void* d_out, int out_size, void* d_ws, size_t ws_size,
                              hipStream_t stream) {
    const float* pi     = (const float*)d_in[0];   // (B, 64)
    const float* mu     = (const float*)d_in[1];   // (64, 8)
    const float* Lc     = (const float*)d_in[2];   // (64, 36)
    const float* target = (const float*)d_in[3];   // (B, 8)
    float* out = (float*)d_out;

    const int B    = in_sizes[0] / KCOMP;             // 131072
    const int nblk = B / (16 * TILES_PER_BLK);        // 1024

    float* W        = (float*)d_ws;                   // 64*48 floats
    float* partials = W + KCOMP * FDIM;               // nblk floats

    gmdn_prep_kernel<<<1, 64, 0, stream>>>(mu, Lc, W);
    gmdn_main_kernel<<<nblk, 128, 0, stream>>>(pi, target, W, partials);
    gmdn_reduce_kernel<<<1, 256, 0, stream>>>(partials, nblk, 1.0f / (float)B, out);
}